// Involution_64802466562135
// MI455X (gfx1250) — compile-verified
//
#include <hip/hip_runtime.h>

// ---- problem constants (B=2, C=256, H=W=56, Cr=64, K=7, G=16) ----
#define HWSZ   3136      // 56*56
#define CIN    256
#define CR     64
#define KKG    784       // 49*16
#define NBATCH 2
#define NTIL   196       // 3136/16 column tiles
#define QSZ    50176     // 56*56*16  (kk stride in raw-view land)
#define OUTPB  802816    // 256*3136 per-batch output elems

// workspace layout (floats): [0..63]=sum [64..127]=sumsq [128..191]=aCoef [192..255]=bCoef
// [256 ...] = t buffer (B * 64 * 3136)
#define WS_T   256

typedef float v2f __attribute__((ext_vector_type(2)));
typedef float v8f __attribute__((ext_vector_type(8)));

// ---------------- kernel 1: t = w1 @ x + b1, plus BN sum/sumsq ----------------
__global__ void __launch_bounds__(256)
k1_gemm1_stats(const float* __restrict__ x, const float* __restrict__ w1,
               const float* __restrict__ b1, float* __restrict__ ws) {
  const int wave = threadIdx.x >> 5;
  const int lane = threadIdx.x & 31;
  int tile = blockIdx.x * 8 + wave;            // 196 blocks * 8 waves = 1568 tiles exact
  const int perB = 4 * NTIL;                   // 4 row tiles * 196 col tiles
  int b  = tile / perB;
  int r  = tile % perB;
  int m0 = (r / NTIL) * 16;
  int n0 = (r % NTIL) * 16;

  const int l15  = lane & 15;
  const int kOff = (lane >> 4) << 1;           // 0 or 2 : K sub-lane offset
  const float* xb   = x + b * (CIN * HWSZ);
  const float* wrow = w1 + (m0 + l15) * CIN + kOff;     // A: lane = row M
  const float* xcol = xb + kOff * HWSZ + (n0 + l15);    // B: lane = col N

  v8f acc = {};
  #pragma unroll 8
  for (int kk = 0; kk < CIN; kk += 4) {
    v2f a = *(const v2f*)(wrow + kk);          // A[M][kk+kOff], A[M][kk+kOff+1]
    v2f bf;
    bf.x = xcol[kk * HWSZ];                    // X[kk+kOff  ][N]
    bf.y = xcol[kk * HWSZ + HWSZ];             // X[kk+kOff+1][N]
    acc = __builtin_amdgcn_wmma_f32_16x16x4_f32(false, a, false, bf,
                                                (short)0, acc, false, false);
  }

  const int halfOff = (lane >> 4) * 8;
  const int col = n0 + l15;
  float* tb   = ws + WS_T + b * (CR * HWSZ);
  float* sumB = ws;
  float* sqB  = ws + 64;
  #pragma unroll
  for (int v = 0; v < 8; ++v) {
    int row = m0 + v + halfOff;
    float val = acc[v] + b1[row];
    tb[row * HWSZ + col] = val;
    // reduce across the 16 lanes that share this row (xor masks <16 keep halves separate)
    float s = val, q = val * val;
    #pragma unroll
    for (int off = 8; off >= 1; off >>= 1) {
      s += __shfl_xor(s, off);
      q += __shfl_xor(q, off);
    }
    if (l15 == 0) {
      atomicAdd(&sumB[row], s);
      atomicAdd(&sqB[row],  q);
    }
  }
}

// ---------------- kernel 2: fold BN stats into per-channel scale/shift ----------------
__global__ void k2_bnprep(const float* __restrict__ gamma,
                          const float* __restrict__ beta,
                          float* __restrict__ ws) {
  int c = threadIdx.x;
  if (c < CR) {
    const float inv_count = 1.0f / (float)(NBATCH * HWSZ);
    float mean = ws[c] * inv_count;
    float var  = ws[64 + c] * inv_count - mean * mean;
    float a    = gamma[c] * rsqrtf(var + 1e-5f);
    ws[128 + c] = a;
    ws[192 + c] = beta[c] - mean * a;
  }
}

// ---------------- kernel 3: k = w2 @ relu(a*t + b) + b2  -> second output region ----------------
__global__ void __launch_bounds__(256)
k3_gemm2(const float* __restrict__ w2, const float* __restrict__ b2,
         const float* __restrict__ ws, float* __restrict__ kout) {
  __shared__ float sAB[128];                   // [0..63]=scale, [64..127]=shift
  if (threadIdx.x < 128) sAB[threadIdx.x] = ws[128 + threadIdx.x];
  __syncthreads();

  const int wave = threadIdx.x >> 5;
  const int lane = threadIdx.x & 31;
  int tile = blockIdx.x * 8 + wave;            // 2401 blocks * 8 = 19208 tiles exact
  const int perB = 49 * NTIL;
  int b  = tile / perB;
  int r  = tile % perB;
  int m0 = (r / NTIL) * 16;
  int n0 = (r % NTIL) * 16;

  const int l15  = lane & 15;
  const int kOff = (lane >> 4) << 1;
  const float* tb   = ws + WS_T + b * (CR * HWSZ);
  const float* wrow = w2 + (m0 + l15) * CR + kOff;
  const float* tcol = tb + kOff * HWSZ + (n0 + l15);

  v8f acc = {};
  #pragma unroll
  for (int kk = 0; kk < CR; kk += 4) {
    v2f a = *(const v2f*)(wrow + kk);
    int k0 = kk + kOff;
    v2f bf;
    bf.x = fmaxf(tcol[kk * HWSZ]        * sAB[k0]     + sAB[64 + k0],     0.0f);
    bf.y = fmaxf(tcol[kk * HWSZ + HWSZ] * sAB[k0 + 1] + sAB[64 + k0 + 1], 0.0f);
    acc = __builtin_amdgcn_wmma_f32_16x16x4_f32(false, a, false, bf,
                                                (short)0, acc, false, false);
  }

  const int halfOff = (lane >> 4) * 8;
  const int col = n0 + l15;
  float* kb = kout + b * (KKG * HWSZ);
  #pragma unroll
  for (int v = 0; v < 8; ++v) {
    int row = m0 + v + halfOff;
    kb[row * HWSZ + col] = acc[v] + b2[row];
  }
}

// ---------------- kernel 4: involution gather, strength-reduced index algebra ----------------
// For a fixed output element (c1, q):  flatP(kk2) = c1*2458624 + kk2*50176 + q.
// Both strides are multiples of 3136, so (ii,jj) = divmod(q % 3136, 56) is tap-invariant,
// the unfold tap index kk advances by +16 (mod 49) per step, and each wrap bumps the
// input channel by one (xoff += 3136). No per-tap division chains remain.
__global__ void __launch_bounds__(256)
k4_involution(const float* __restrict__ x, const float* __restrict__ kbuf,
              float* __restrict__ out) {
  int idx = blockIdx.x * blockDim.x + threadIdx.x;  // 6272*256 = 1,605,632 exact
  int b  = idx / OUTPB;
  int of = idx % OUTPB;
  int c1 = of / QSZ;                // outer group-channel
  int q  = of % QSZ;                // i*896 + j*16 + g

  int kk   = q / HWSZ;              // initial tap index (0..15)
  int rem2 = q - kk * HWSZ;
  int ii   = rem2 / 56;
  int jj   = rem2 - ii * 56;

  const float* kb = kbuf + (size_t)b * (KKG * HWSZ) + q;           // + kk2*QSZ per tap
  // base points at x[b][c1*16][ii-3][jj-3]; taps add xoff + di*56 + dj (guarded)
  const float* xbase = x + (size_t)b * (CIN * HWSZ) + c1 * 16 * HWSZ
                         + (ii - 3) * 56 + (jj - 3);
  int xoff = 0;

  float acc = 0.0f;
  #pragma unroll
  for (int kk2 = 0; kk2 < 49; ++kk2) {
    float kval = kb[kk2 * QSZ];                  // immediate-offset loads off one base
    int di = (kk * 37) >> 8;                     // exact kk/7 for kk < 64
    int dj = kk - 7 * di;
    float xv = 0.0f;
    if ((unsigned)(ii + di - 3) < 56u && (unsigned)(jj + dj - 3) < 56u)
      xv = xbase[xoff + di * 56 + dj];
    acc = fmaf(kval, xv, acc);
    kk += 16;
    if (kk >= 49) { kk -= 49; xoff += HWSZ; }    // channel wrap: one cond-subtract
  }
  out[idx] = acc;
}

// ---------------- host-side launch ----------------
extern "C" void kernel_launch(void* const* d_in, const int* in_sizes, int n_in,
                              void* d_out, int out_size, void* d_ws, size_t ws_size,
                              hipStream_t stream) {
  const float* x     = (const float*)d_in[0];
  const float* w1    = (const float*)d_in[1];
  const float* b1    = (const float*)d_in[2];
  const float* gamma = (const float*)d_in[3];
  const float* beta  = (const float*)d_in[4];
  const float* w2    = (const float*)d_in[5];
  const float* b2    = (const float*)d_in[6];

  float* out  = (float*)d_out;
  float* kout = out + (size_t)NBATCH * OUTPB;   // second tuple element lives here
  float* ws   = (float*)d_ws;                   // 256 + 2*64*3136 floats ≈ 1.6 MB

  hipMemsetAsync(ws, 0, 128 * sizeof(float), stream);   // zero BN sum/sumsq
  k1_gemm1_stats<<<196, 256, 0, stream>>>(x, w1, b1, ws);
  k2_bnprep<<<1, 64, 0, stream>>>(gamma, beta, ws);
  k3_gemm2<<<2401, 256, 0, stream>>>(w2, b2, ws, kout);
  k4_involution<<<6272, 256, 0, stream>>>(x, kout, out);
}